// RMB_10823317586415
// MI455X (gfx1250) — compile-verified
//
#include <hip/hip_runtime.h>
#include <math.h>

typedef float v2f __attribute__((ext_vector_type(2)));
typedef float v8f __attribute__((ext_vector_type(8)));

__device__ __forceinline__ v8f wmma4(v8f c, v2f a, v2f b) {
  // D = A(16x4) x B(4x16) + C(16x16), fp32
  return __builtin_amdgcn_wmma_f32_16x16x4_f32(false, a, false, b, (short)0, c, false, false);
}
__device__ __forceinline__ float lrelu(float x) { return x > 0.f ? x : 0.1f * x; }
__device__ __forceinline__ float gelu_(float x) { return 0.5f * x * (1.f + erff(x * 0.70710678118654752f)); }

// ---------------- workspace layout (float offsets) ----------------
constexpr int SZ_TROWF = 128 * 160;          // forward row-DFT table (w x [cos|sin])
constexpr int SZ_TCOL  = 2 * 128 * 256;      // col-DFT tables re/im, K over stacked [R;I]
constexpr int SZ_TROWI = 160 * 128;          // inverse row-DFT table
constexpr int OFF_TROWF = 0;
constexpr int OFF_TCOLF = OFF_TROWF + SZ_TROWF;
constexpr int OFF_TCOLI = OFF_TCOLF + SZ_TCOL;
constexpr int OFF_TROWI = OFF_TCOLI + SZ_TCOL;
constexpr int OFF_E1W1P = OFF_TROWI + SZ_TROWI;     // e1 w1 padded 128x112
constexpr int OFF_E1B1P = OFF_E1W1P + 128 * 112;
constexpr int OFF_E1W2P = OFF_E1B1P + 112;          // e1 w2 padded 112x128
constexpr int OFF_E2W1P = OFF_E1W2P + 112 * 128;    // e2 w1 padded 128x160
constexpr int OFF_E2B1P = OFF_E2W1P + 128 * 160;
constexpr int OFF_E2W2P = OFF_E2B1P + 160;          // e2 w2 padded 160x128
constexpr int TAB_TOTAL = OFF_E2W2P + 160 * 128;    // 241936 entries
constexpr int PLANE_F   = 20480;                    // 2*128*80 == 128*160 floats per plane
constexpr int NPLANE    = 512;                      // B*C = 8*64
constexpr int OFF_Y  = ((TAB_TOTAL + 63) / 64) * 64;
constexpr int OFF_S  = OFF_Y + NPLANE * PLANE_F;    // also reused as Z (inv-col output)
constexpr int OFF_A  = OFF_S + NPLANE * PLANE_F;    // magnitude, (b,h,80,64) token layout
constexpr int OFF_P  = OFF_A + 8 * 128 * 80 * 64;   // phase
constexpr int OFF_P1 = OFF_P + 8 * 128 * 80 * 64;   // conv scratch
constexpr int OFF_FB = OFF_P1 + 8 * 128 * 80 * 64;  // FB-branch output (B,H,W,64)

// ---------------- table + weight-padding setup ----------------
__global__ void k_tables(float* ws,
                         const float* e1w1, const float* e1b1, const float* e1w2,
                         const float* e2w1, const float* e2b1, const float* e2w2) {
  int i = blockIdx.x * 256 + threadIdx.x;
  const float W128 = 6.28318530717958647692f / 128.f;
  if (i < SZ_TROWF) {  // forward row rDFT: [w][n], n<80 cos, n>=80 sin, k>=65 pad 0
    int w = i / 160, n = i % 160, k = (n < 80) ? n : n - 80;
    float v = 0.f;
    if (k < 65) { int m = (w * k) & 127; float th = -W128 * (float)m; v = (n < 80) ? cosf(th) : sinf(th); }
    ws[OFF_TROWF + i] = v; return;
  }
  i -= SZ_TROWF;
  if (i < SZ_TCOL) {  // forward col DFT: part0=Re rows, part1=Im rows; K'=[h | 128+h]
    int part = i / 32768, j = i % 32768, hk = j / 256, kp = j % 256, h = kp & 127;
    int m = (hk * h) & 127; float th = -W128 * (float)m; float c = cosf(th), s = sinf(th);
    ws[OFF_TCOLF + i] = (part == 0) ? (kp < 128 ? c : -s) : (kp < 128 ? s : c); return;
  }
  i -= SZ_TCOL;
  if (i < SZ_TCOL) {  // inverse col DFT (scaled 1/128)
    int part = i / 32768, j = i % 32768, h = j / 256, kp = j % 256, hk = kp & 127;
    int m = (h * hk) & 127; float th = W128 * (float)m;
    float c = cosf(th) * 0.0078125f, s = sinf(th) * 0.0078125f;
    ws[OFF_TCOLI + i] = (part == 0) ? (kp < 128 ? c : -s) : (kp < 128 ? s : c); return;
  }
  i -= SZ_TCOL;
  if (i < SZ_TROWI) {  // inverse row rDFT with Hermitian weights: [kp][w]
    int kp = i / 128, w = i % 128, k = (kp < 80) ? kp : kp - 80;
    float v = 0.f;
    if (k < 65) {
      float wk = (k == 0 || k == 64) ? 1.f : 2.f;
      int m = (w * k) & 127; float th = W128 * (float)m;
      v = ((kp < 80) ? wk * cosf(th) : -wk * sinf(th)) * 0.0078125f;
    }
    ws[OFF_TROWI + i] = v; return;
  }
  i -= SZ_TROWI;
  if (i < 128 * 112) { int k = i / 112, n = i % 112; ws[OFF_E1W1P + i] = (n < 102) ? e1w1[k * 102 + n] : 0.f; return; }
  i -= 128 * 112;
  if (i < 112) { ws[OFF_E1B1P + i] = (i < 102) ? e1b1[i] : 0.f; return; }
  i -= 112;
  if (i < 112 * 128) { int k = i / 128, n = i % 128; ws[OFF_E1W2P + i] = (k < 102) ? e1w2[k * 128 + n] : 0.f; return; }
  i -= 112 * 128;
  if (i < 128 * 160) { int k = i / 160, n = i % 160; ws[OFF_E2W1P + i] = (n < 153) ? e2w1[k * 153 + n] : 0.f; return; }
  i -= 128 * 160;
  if (i < 160) { ws[OFF_E2B1P + i] = (i < 153) ? e2b1[i] : 0.f; return; }
  i -= 160;
  if (i < 160 * 128) { int k = i / 128, n = i % 128; ws[OFF_E2W2P + i] = (k < 153) ? e2w2[k * 128 + n] : 0.f; return; }
}

// ---------------- forward row rDFT (GEMM per (b,c) plane) ----------------
__global__ void k_rowfft_f(const float* __restrict__ x, const float* __restrict__ trf,
                           float* __restrict__ Y) {
  int lane = threadIdx.x & 31, wid = threadIdx.x >> 5;
  int job = blockIdx.x * 4 + wid;                 // 512 planes * 8 Mtiles * 10 Ntiles
  int plane = job / 80, t = job % 80, mt = t / 10, nt = t % 10;
  int b = plane >> 6, c = plane & 63;
  int mrow = lane & 15, koff = (lane >> 4) * 2, col = lane & 15, hi8 = (lane >> 4) * 8;
  int h = mt * 16 + mrow, n = nt * 16 + col;
  const float* Ab = x + ((size_t)(b * 128 + h) * 128) * 64 + c;
  __builtin_prefetch(trf + n, 0, 1);              // twiddle table stays hot in L2/WGP$
  v8f acc = {};
  for (int k0 = 0; k0 < 128; k0 += 4) {
    v2f a; a.x = Ab[(size_t)(k0 + koff) * 64]; a.y = Ab[(size_t)(k0 + koff + 1) * 64];
    v2f bb; bb.x = trf[(k0 + koff) * 160 + n]; bb.y = trf[(k0 + koff + 1) * 160 + n];
    acc = wmma4(acc, a, bb);
  }
  float* Yp = Y + plane * PLANE_F;
#pragma unroll
  for (int v = 0; v < 8; v++) {
    int hh = mt * 16 + v + hi8;
    int off = (n < 80) ? (hh * 80 + n) : (10240 + hh * 80 + (n - 80));
    Yp[off] = acc[v];
  }
}

// ---------------- forward column DFT ----------------
__global__ void k_colfft_f(const float* __restrict__ Y, const float* __restrict__ tcolf,
                           float* __restrict__ S) {
  int lane = threadIdx.x & 31, wid = threadIdx.x >> 5;
  int job = blockIdx.x * 4 + wid;                 // 512 * (2 parts * 8 M * 5 N)
  int plane = job / 80, t = job % 80, part = t / 40, tt = t % 40, mt = tt / 5, nt = tt % 5;
  int mrow = lane & 15, koff = (lane >> 4) * 2, col = lane & 15, hi8 = (lane >> 4) * 8;
  int hk = mt * 16 + mrow, n = nt * 16 + col;
  const float* At = tcolf + part * 32768 + hk * 256;
  const float* Bp = Y + plane * PLANE_F;
  __builtin_prefetch(At, 0, 1);
  v8f acc = {};
  for (int k0 = 0; k0 < 256; k0 += 4) {
    v2f a = *(const v2f*)(At + k0 + koff);
    v2f bb; bb.x = Bp[(k0 + koff) * 80 + n]; bb.y = Bp[(k0 + koff + 1) * 80 + n];
    acc = wmma4(acc, a, bb);
  }
  float* Sp = S + plane * PLANE_F + part * 10240;
#pragma unroll
  for (int v = 0; v < 8; v++) Sp[(mt * 16 + v + hi8) * 80 + n] = acc[v];
}

// ---------------- abs/angle + transpose to token layout ----------------
__global__ void k_absangle(const float* __restrict__ S, float* __restrict__ Aarr,
                           float* __restrict__ Parr) {
  int i = blockIdx.x * 256 + threadIdx.x;          // 512*128*80
  int plane = i / 10240, r = i % 10240, k = r % 80, hk = r / 80;
  int b = plane >> 6, c = plane & 63;
  size_t dst = ((size_t)((b * 128 + hk) * 80 + k)) * 64 + c;
  if (k < 65) {
    float re = S[plane * PLANE_F + r] + 1e-8f;
    float im = S[plane * PLANE_F + 10240 + r];
    Aarr[dst] = sqrtf(re * re + im * im);
    Parr[dst] = atan2f(im, re);
  } else { Aarr[dst] = 0.f; Parr[dst] = 0.f; }
}

// ---------------- magnitude branch: fused 1x1 -> leaky -> 1x1 -> leaky ----------------
__global__ void k_conv1x1a(float* __restrict__ Aarr,
                           const float* __restrict__ w1, const float* __restrict__ b1,
                           const float* __restrict__ w2, const float* __restrict__ b2) {
  __shared__ float t[16 * 68];
  int lane = threadIdx.x & 31, wid = threadIdx.x >> 5;
  int mrow = lane & 15, koff = (lane >> 4) * 2, col = lane & 15, hi8 = (lane >> 4) * 8;
  int tile = blockIdx.x, n = wid * 16 + col;
  int q = tile * 16 + mrow, b = q / 8320, r = q % 8320, h = r / 65, w = r % 65;
  const float* Ab = Aarr + ((size_t)((b * 128 + h) * 80 + w)) * 64;
  __builtin_prefetch(w1 + n * 64, 0, 1);
  v8f acc = {};
  for (int k0 = 0; k0 < 64; k0 += 4) {
    v2f a = *(const v2f*)(Ab + k0 + koff);
    v2f bb = *(const v2f*)(w1 + n * 64 + k0 + koff);   // w1[oc][ic]
    acc = wmma4(acc, a, bb);
  }
  float bs = b1[n];
#pragma unroll
  for (int v = 0; v < 8; v++) t[(v + hi8) * 68 + n] = lrelu(acc[v] + bs);
  __syncthreads();
  v8f acc2 = {};
  for (int k0 = 0; k0 < 64; k0 += 4) {
    v2f a = *(const v2f*)&t[mrow * 68 + k0 + koff];
    v2f bb = *(const v2f*)(w2 + n * 64 + k0 + koff);
    acc2 = wmma4(acc2, a, bb);
  }
  float bs2 = b2[n];
#pragma unroll
  for (int v = 0; v < 8; v++) {
    int m = v + hi8, qq = tile * 16 + m, bb2 = qq / 8320, rr = qq % 8320;
    int hh = rr / 65, ww = rr % 65;
    Aarr[((size_t)((bb2 * 128 + hh) * 80 + ww)) * 64 + n] = lrelu(acc2[v] + bs2);
  }
}

// ---------------- phase branch: 3x3 conv as implicit GEMM (+bias+leaky) ----------------
// Halo tiles are staged with GLOBAL_LOAD_ASYNC_TO_LDS (ASYNCcnt path) once the
// previous tap's WMMA chain has passed the barrier (single buffer); border lanes write LDS
// zeros through the normal DS path. Each wave drains its own ASYNCcnt before
// the barrier, so post-barrier all staging is visible workgroup-wide.
__global__ void k_conv3(const float* __restrict__ in, float* __restrict__ out,
                        const float* __restrict__ w, const float* __restrict__ bias) {
  __shared__ float As[16 * 68];
  int tid = threadIdx.x, lane = tid & 31, wid = tid >> 5;
  int mrow = lane & 15, koff = (lane >> 4) * 2, col = lane & 15, hi8 = (lane >> 4) * 8;
  int tile = blockIdx.x, n = wid * 16 + col;
  v8f acc = {};
  for (int tap = 0; tap < 9; tap++) {
    int dy = tap / 3 - 1, dx = tap % 3 - 1;
    for (int i = tid; i < 1024; i += 128) {
      int m = i >> 6, ic = i & 63;
      int q = tile * 16 + m, b = q / 8320, r = q % 8320, h = r / 65, ww = r % 65;
      int hh = h + dy, w2 = ww + dx;
      unsigned ldsoff = (unsigned)((m * 68 + ic) * 4);
      if (hh >= 0 && hh < 128 && w2 >= 0 && w2 < 65) {
        const float* gp = in + ((size_t)((b * 128 + hh) * 80 + w2)) * 64 + ic;
        asm volatile("global_load_async_to_lds_b32 %0, %1, off"
                     :: "v"(ldsoff), "v"(gp) : "memory");
      } else {
        As[m * 68 + ic] = 0.f;
      }
    }
    asm volatile("s_wait_asynccnt 0x0" ::: "memory");
    __syncthreads();
    for (int k0 = 0; k0 < 64; k0 += 4) {
      v2f a = *(const v2f*)&As[mrow * 68 + k0 + koff];
      v2f bb;
      bb.x = w[(size_t)(n * 64 + k0 + koff) * 9 + tap];     // OIHW
      bb.y = w[(size_t)(n * 64 + k0 + koff + 1) * 9 + tap];
      acc = wmma4(acc, a, bb);
    }
    __syncthreads();
  }
  float bs = bias[n];
#pragma unroll
  for (int v = 0; v < 8; v++) {
    int m = v + hi8, q = tile * 16 + m, b = q / 8320, r = q % 8320, h = r / 65, ww = r % 65;
    out[((size_t)((b * 128 + h) * 80 + ww)) * 64 + n] = lrelu(acc[v] + bs);
  }
}

// ---------------- reconstruct complex spectrum ----------------
__global__ void k_recon(const float* __restrict__ Aarr, const float* __restrict__ Parr,
                        float* __restrict__ S2) {
  int i = blockIdx.x * 256 + threadIdx.x;
  int plane = i / 10240, r = i % 10240, k = r % 80, hk = r / 80;
  int b = plane >> 6, c = plane & 63;
  float re = 0.f, im = 0.f;
  if (k < 65) {
    size_t src = ((size_t)((b * 128 + hk) * 80 + k)) * 64 + c;
    float a = Aarr[src], p = Parr[src], s, co;
    sincosf(p, &s, &co);
    re = a * co + 1e-8f; im = a * s;
  }
  S2[plane * PLANE_F + r] = re;
  S2[plane * PLANE_F + 10240 + r] = im;
}

// ---------------- inverse column DFT ----------------
__global__ void k_colfft_i(const float* __restrict__ S2, const float* __restrict__ tcoli,
                           float* __restrict__ Z) {
  int lane = threadIdx.x & 31, wid = threadIdx.x >> 5;
  int job = blockIdx.x * 4 + wid;                 // 512 * 8 M * 10 Nvirt
  int plane = job / 80, t = job % 80, mt = t / 10, nt = t % 10;
  int part = (nt >= 5) ? 1 : 0;
  int mrow = lane & 15, koff = (lane >> 4) * 2, col = lane & 15, hi8 = (lane >> 4) * 8;
  int h = mt * 16 + mrow;
  int nb = ((part ? nt - 5 : nt) * 16) + col;
  const float* At = tcoli + part * 32768 + h * 256;
  const float* Bp = S2 + plane * PLANE_F;
  __builtin_prefetch(At, 0, 1);
  v8f acc = {};
  for (int k0 = 0; k0 < 256; k0 += 4) {
    v2f a = *(const v2f*)(At + k0 + koff);
    v2f bb; bb.x = Bp[(k0 + koff) * 80 + nb]; bb.y = Bp[(k0 + koff + 1) * 80 + nb];
    acc = wmma4(acc, a, bb);
  }
  float* Zp = Z + plane * PLANE_F;
  int nout = nt * 16 + col;
#pragma unroll
  for (int v = 0; v < 8; v++) Zp[(mt * 16 + v + hi8) * 160 + nout] = acc[v];
}

// ---------------- inverse row rDFT + abs + skip ----------------
__global__ void k_rowfft_i(const float* __restrict__ Z, const float* __restrict__ tri,
                           const float* __restrict__ x, float* __restrict__ FB) {
  int lane = threadIdx.x & 31, wid = threadIdx.x >> 5;
  int job = blockIdx.x * 4 + wid;                 // 512 * 8 M * 8 N
  int plane = job / 64, t = job % 64, mt = t / 8, nt = t % 8;
  int b = plane >> 6, c = plane & 63;
  int mrow = lane & 15, koff = (lane >> 4) * 2, col = lane & 15, hi8 = (lane >> 4) * 8;
  int h = mt * 16 + mrow, n = nt * 16 + col;
  const float* Ap = Z + plane * PLANE_F + h * 160;
  __builtin_prefetch(tri + n, 0, 1);
  v8f acc = {};
  for (int k0 = 0; k0 < 160; k0 += 4) {
    v2f a = *(const v2f*)(Ap + k0 + koff);
    v2f bb; bb.x = tri[(k0 + koff) * 128 + n]; bb.y = tri[(k0 + koff + 1) * 128 + n];
    acc = wmma4(acc, a, bb);
  }
#pragma unroll
  for (int v = 0; v < 8; v++) {
    int hh = mt * 16 + v + hi8;
    size_t ix = ((size_t)((b * 128 + hh) * 128 + n)) * 64 + c;
    FB[ix] = fabsf(acc[v] + 1e-8f) + 1e-8f + x[ix];
  }
}

// ---------------- fully fused MoE over 32-token tiles ----------------
__global__ void k_moe(const float* __restrict__ FB,
                      const float* __restrict__ wp, const float* __restrict__ bp,
                      const float* __restrict__ e0w1, const float* __restrict__ e0b1,
                      const float* __restrict__ e0w2, const float* __restrict__ e0b2,
                      const float* __restrict__ e1w1, const float* __restrict__ e1b1,
                      const float* __restrict__ e1w2, const float* __restrict__ e1b2,
                      const float* __restrict__ e2w1, const float* __restrict__ e2b1,
                      const float* __restrict__ e2w2, const float* __restrict__ e2b2,
                      const float* __restrict__ wg, const float* __restrict__ bg,
                      const float* __restrict__ wmix, const float* __restrict__ bmix,
                      const float* __restrict__ wout, const float* __restrict__ bout,
                      float* __restrict__ out) {
  __shared__ float hdn[32 * 132];
  __shared__ float t1[32 * 164];
  __shared__ float fus[32 * 132];
  __shared__ float alph[32 * 3];
  int tid = threadIdx.x, lane = tid & 31, wid = tid >> 5;
  int mrow = lane & 15, koff = (lane >> 4) * 2, col = lane & 15, hi8 = (lane >> 4) * 8;
  int tok0 = blockIdx.x * 32;

  // 1) hdn = x @ wp_in + bp_in   (32x64 @ 64x128)
  for (int job = wid; job < 16; job += 4) {
    int mt = job >> 3, nt = job & 7, n = nt * 16 + col;
    const float* Ab = FB + (size_t)(tok0 + mt * 16 + mrow) * 64;
    v8f acc = {};
    for (int k0 = 0; k0 < 64; k0 += 4) {
      v2f a = *(const v2f*)(Ab + k0 + koff);
      v2f bb; bb.x = wp[(k0 + koff) * 128 + n]; bb.y = wp[(k0 + koff + 1) * 128 + n];
      acc = wmma4(acc, a, bb);
    }
    float bs = bp[n];
#pragma unroll
    for (int v = 0; v < 8; v++) hdn[(mt * 16 + v + hi8) * 132 + n] = acc[v] + bs;
  }
  __syncthreads();

  // 2) gating: top-2 of 3 softmax
  if (tid < 32) {
    const float* hr = &hdn[tid * 132];
    float l0 = bg[0], l1 = bg[1], l2 = bg[2];
    for (int k = 0; k < 128; k++) {
      float h = hr[k];
      l0 += h * wg[k * 3]; l1 += h * wg[k * 3 + 1]; l2 += h * wg[k * 3 + 2];
    }
    const float invT = 1.f / (1.f + 1e-8f);
    l0 *= invT; l1 *= invT; l2 *= invT;
    float hi = fmaxf(l0, fmaxf(l1, l2));
    float lo = fminf(l0, fminf(l1, l2));
    float mid = l0 + l1 + l2 - hi - lo;  // 2nd largest
    float a0 = (l0 >= mid) ? expf(l0 - hi) : 0.f;
    float a1 = (l1 >= mid) ? expf(l1 - hi) : 0.f;
    float a2 = (l2 >= mid) ? expf(l2 - hi) : 0.f;
    float s = a0 + a1 + a2;
    alph[tid * 3 + 0] = a0 / s; alph[tid * 3 + 1] = a1 / s; alph[tid * 3 + 2] = a2 / s;
  }
  for (int i = tid; i < 32 * 132; i += 128) fus[i] = 0.f;
  __syncthreads();

  // 3) experts (hidden padded to 16-multiples; pads are exact zeros)
  for (int e = 0; e < 3; e++) {
    const float *W1, *B1, *W2, *B2; int EH;
    if (e == 0)      { W1 = e0w1; B1 = e0b1; W2 = e0w2; B2 = e0b2; EH = 128; }
    else if (e == 1) { W1 = e1w1; B1 = e1b1; W2 = e1w2; B2 = e1b2; EH = 112; }
    else             { W1 = e2w1; B1 = e2b1; W2 = e2w2; B2 = e2b2; EH = 160; }
    int ntE = EH >> 4;
    __builtin_prefetch(W1, 0, 1);
    __builtin_prefetch(W2, 0, 1);
    for (int job = wid; job < 2 * ntE; job += 4) {
      int mt = job / ntE, nt = job % ntE, n = nt * 16 + col;
      v8f acc = {};
      for (int k0 = 0; k0 < 128; k0 += 4) {
        v2f a = *(const v2f*)&hdn[(mt * 16 + mrow) * 132 + k0 + koff];
        v2f bb; bb.x = W1[(k0 + koff) * EH + n]; bb.y = W1[(k0 + koff + 1) * EH + n];
        acc = wmma4(acc, a, bb);
      }
      float bs = B1[n];
#pragma unroll
      for (int v = 0; v < 8; v++) t1[(mt * 16 + v + hi8) * 164 + n] = gelu_(acc[v] + bs);
    }
    __syncthreads();
    for (int job = wid; job < 16; job += 4) {
      int mt = job >> 3, nt = job & 7, n = nt * 16 + col;
      v8f acc = {};
      for (int k0 = 0; k0 < EH; k0 += 4) {
        v2f a = *(const v2f*)&t1[(mt * 16 + mrow) * 164 + k0 + koff];
        v2f bb; bb.x = W2[(k0 + koff) * 128 + n]; bb.y = W2[(k0 + koff + 1) * 128 + n];
        acc = wmma4(acc, a, bb);
      }
      float bs = B2[n];
#pragma unroll
      for (int v = 0; v < 8; v++) {
        int m = mt * 16 + v + hi8;
        fus[m * 132 + n] += alph[m * 3 + e] * (acc[v] + bs);
      }
    }
    __syncthreads();
  }

  // 4) mixed = gelu(fused @ wmix + bmix)  -> t1
  for (int job = wid; job < 16; job += 4) {
    int mt = job >> 3, nt = job & 7, n = nt * 16 + col;
    v8f acc = {};
    for (int k0 = 0; k0 < 128; k0 += 4) {
      v2f a = *(const v2f*)&fus[(mt * 16 + mrow) * 132 + k0 + koff];
      v2f bb; bb.x = wmix[(k0 + koff) * 128 + n]; bb.y = wmix[(k0 + koff + 1) * 128 + n];
      acc = wmma4(acc, a, bb);
    }
    float bs = bmix[n];
#pragma unroll
    for (int v = 0; v < 8; v++) t1[(mt * 16 + v + hi8) * 164 + n] = gelu_(acc[v] + bs);
  }
  __syncthreads();

  // 5) out = (hdn + mixed) @ wout + bout
  for (int job = wid; job < 8; job += 4) {
    int mt = job >> 2, nt = job & 3, n = nt * 16 + col;
    v8f acc = {};
    for (int k0 = 0; k0 < 128; k0 += 4) {
      v2f a1 = *(const v2f*)&hdn[(mt * 16 + mrow) * 132 + k0 + koff];
      v2f a2 = *(const v2f*)&t1[(mt * 16 + mrow) * 164 + k0 + koff];
      v2f a = a1 + a2;
      v2f bb; bb.x = wout[(k0 + koff) * 64 + n]; bb.y = wout[(k0 + koff + 1) * 64 + n];
      acc = wmma4(acc, a, bb);
    }
    float bs = bout[n];
#pragma unroll
    for (int v = 0; v < 8; v++)
      out[(size_t)(tok0 + mt * 16 + v + hi8) * 64 + n] = acc[v] + bs;
  }
}

// ---------------- host launch ----------------
extern "C" void kernel_launch(void* const* d_in, const int* in_sizes, int n_in,
                              void* d_out, int out_size, void* d_ws, size_t ws_size,
                              hipStream_t stream) {
  (void)in_sizes; (void)n_in; (void)out_size; (void)ws_size;
  const float* x     = (const float*)d_in[0];
  const float* s1w1  = (const float*)d_in[1];  const float* s1b1 = (const float*)d_in[2];
  const float* s1w2  = (const float*)d_in[3];  const float* s1b2 = (const float*)d_in[4];
  const float* s2w1  = (const float*)d_in[5];  const float* s2b1 = (const float*)d_in[6];
  const float* s2w2  = (const float*)d_in[7];  const float* s2b2 = (const float*)d_in[8];
  const float* wp    = (const float*)d_in[9];  const float* bpn  = (const float*)d_in[10];
  const float* e0w1  = (const float*)d_in[11]; const float* e0b1 = (const float*)d_in[12];
  const float* e0w2  = (const float*)d_in[13]; const float* e0b2 = (const float*)d_in[14];
  const float* e1w1  = (const float*)d_in[15]; const float* e1b1 = (const float*)d_in[16];
  const float* e1w2  = (const float*)d_in[17]; const float* e1b2 = (const float*)d_in[18];
  const float* e2w1  = (const float*)d_in[19]; const float* e2b1 = (const float*)d_in[20];
  const float* e2w2  = (const float*)d_in[21]; const float* e2b2 = (const float*)d_in[22];
  const float* wg    = (const float*)d_in[23]; const float* bg   = (const float*)d_in[24];
  const float* wmix  = (const float*)d_in[25]; const float* bmix = (const float*)d_in[26];
  const float* wout  = (const float*)d_in[27]; const float* bout = (const float*)d_in[28];
  float* outp = (float*)d_out;
  float* ws = (float*)d_ws;

  float* trowf = ws + OFF_TROWF;  float* tcolf = ws + OFF_TCOLF;
  float* tcoli = ws + OFF_TCOLI;  float* trowi = ws + OFF_TROWI;
  float* Y  = ws + OFF_Y;   float* S  = ws + OFF_S;
  float* Aa = ws + OFF_A;   float* Pa = ws + OFF_P;  float* P1 = ws + OFF_P1;
  float* FB = ws + OFF_FB;

  k_tables<<<(TAB_TOTAL + 255) / 256, 256, 0, stream>>>(ws, e1w1, e1b1, e1w2, e2w1, e2b1, e2w2);
  k_rowfft_f<<<10240, 128, 0, stream>>>(x, trowf, Y);
  k_colfft_f<<<10240, 128, 0, stream>>>(Y, tcolf, S);
  k_absangle<<<20480, 256, 0, stream>>>(S, Aa, Pa);
  k_conv1x1a<<<4160, 128, 0, stream>>>(Aa, s1w1, s1b1, s1w2, s1b2);
  k_conv3<<<4160, 128, 0, stream>>>(Pa, P1, s2w1, s2b1);
  k_conv3<<<4160, 128, 0, stream>>>(P1, Pa, s2w2, s2b2);
  k_recon<<<20480, 256, 0, stream>>>(Aa, Pa, Y);      // Y reused as recon spectrum
  k_colfft_i<<<10240, 128, 0, stream>>>(Y, tcoli, S); // S reused as Z (h x [re|im])
  k_rowfft_i<<<8192, 128, 0, stream>>>(S, trowi, x, FB);
  k_moe<<<4096, 128, 0, stream>>>(FB, wp, bpn,
                                  e0w1, e0b1, e0w2, e0b2,
                                  ws + OFF_E1W1P, ws + OFF_E1B1P, ws + OFF_E1W2P, e1b2,
                                  ws + OFF_E2W1P, ws + OFF_E2B1P, ws + OFF_E2W2P, e2b2,
                                  wg, bg, wmix, bmix, wout, bout, outp);
}